// LogicLayer_48808008352082
// MI455X (gfx1250) — compile-verified
//
#include <hip/hip_runtime.h>

#define BATCH   4096
#define IN_DIM  8192
#define OUT_DIM 8192
#define ROWS    4          // batch rows per workgroup (128 KiB LDS slab)
#define TPB     256        // 8 waves (wave32)

// ---- Kernel 1: coeffs[j] = softmax(weights[j,:]) @ GATE_COEFFS (float4 per column).
// Gate table entries are 0/±1/±2 -> fold the 16x4 matmul into adds (no constant loads).
__global__ __launch_bounds__(TPB) void gate_coeffs_kernel(
    const float* __restrict__ weights, float4* __restrict__ coeffs) {
    int j = blockIdx.x * blockDim.x + threadIdx.x;
    if (j >= OUT_DIM) return;
    const float* wr = weights + (size_t)j * 16;
    float p[16];
    float m = -3.4e38f;
#pragma unroll
    for (int k = 0; k < 16; ++k) { p[k] = wr[k]; m = fmaxf(m, p[k]); }
    float s = 0.f;
#pragma unroll
    for (int k = 0; k < 16; ++k) { p[k] = expf(p[k] - m); s += p[k]; }
    float inv = 1.f / s;
#pragma unroll
    for (int k = 0; k < 16; ++k) p[k] *= inv;
    // Columns of the gate table:
    float c0  = p[8] + p[9] + p[10] + p[11] + p[12] + p[13] + p[14] + p[15];
    float ca  = p[2] + p[3] + p[6] + p[7] - p[8] - p[9] - p[12] - p[13];
    float cb  = p[4] + p[5] + p[6] + p[7] - p[8] - p[9] - p[10] - p[11];
    float cab = p[1] - p[2] - p[4] - 2.f * p[6] - p[7]
              + p[8] + 2.f * p[9] + p[11] + p[13] - p[14];
    coeffs[j] = make_float4(c0, ca, cb, cab);
}

// ---- async global -> LDS copy of one 16B chunk (gfx1250 ASYNCcnt path).
// lds_escape: flat pointer to the shared array, passed (unused in the template)
// so LLVM sees the LDS region as captured and keeps the post-barrier ds_loads.
__device__ __forceinline__ void async_g2l_b128(unsigned lds_byte_off, const void* gaddr,
                                               void* lds_escape) {
    asm volatile("global_load_async_to_lds_b128 %0, %1, off"
                 :
                 : "v"(lds_byte_off), "v"(gaddr), "v"(lds_escape)
                 : "memory");
}

__device__ __forceinline__ void wait_asynccnt0() {
#if __has_builtin(__builtin_amdgcn_s_wait_asynccnt)
    __builtin_amdgcn_s_wait_asynccnt(0);
#else
    asm volatile("s_wait_asynccnt 0" ::: "memory");
#endif
}

// ---- Kernel 2: main fused gate kernel.
// Each workgroup: ROWS contiguous batch rows of x staged in LDS, all OUT_DIM columns.
__global__ __launch_bounds__(TPB) void logic_layer_kernel(
    const float* __restrict__ x,
    const int* __restrict__ idx_a,
    const int* __restrict__ idx_b,
    const float4* __restrict__ coeffs,
    float* __restrict__ out) {
    __shared__ float smem[ROWS * IN_DIM];   // only __shared__ object -> LDS offset 0

    const int  tid = threadIdx.x;
    const long i0  = (long)blockIdx.x * ROWS;
    const char* gbase = (const char*)(x + i0 * (long)IN_DIM);  // 128 KiB contiguous slab

    // Stage ROWS*IN_DIM floats (131072 B) into LDS: 8192 x 16B chunks, 32 per thread.
    constexpr int CHUNKS = (ROWS * IN_DIM * 4) / (16 * TPB);   // 32
#pragma unroll 8
    for (int c = 0; c < CHUNKS; ++c) {
        unsigned off = (unsigned)(c * TPB + tid) * 16u;
        async_g2l_b128(off, gbase + off, (void*)smem);
    }
    wait_asynccnt0();      // this wave's async LDS writes complete
    __syncthreads();       // all waves' writes complete

    // Column pairs: int2 idx loads, 2x float4 coeffs, LDS gathers, float2 stores
    // (256 B contiguous per wave32 store).
    const int2* __restrict__ ia2 = (const int2*)idx_a;
    const int2* __restrict__ ib2 = (const int2*)idx_b;
#pragma unroll 2
    for (int t = tid; t < OUT_DIM / 2; t += TPB) {
        const int2   ia = ia2[t];
        const int2   ib = ib2[t];
        const float4 cf0 = coeffs[2 * t];
        const float4 cf1 = coeffs[2 * t + 1];
#pragma unroll
        for (int r = 0; r < ROWS; ++r) {
            const float a0 = smem[r * IN_DIM + ia.x];
            const float b0 = smem[r * IN_DIM + ib.x];
            const float a1 = smem[r * IN_DIM + ia.y];
            const float b1 = smem[r * IN_DIM + ib.y];
            float2 o;
            o.x = cf0.x + cf0.y * a0 + cf0.z * b0 + cf0.w * (a0 * b0);
            o.y = cf1.x + cf1.y * a1 + cf1.z * b1 + cf1.w * (a1 * b1);
            *(float2*)&out[(i0 + r) * (long)OUT_DIM + 2 * t] = o;
        }
    }
}

extern "C" void kernel_launch(void* const* d_in, const int* in_sizes, int n_in,
                              void* d_out, int out_size, void* d_ws, size_t ws_size,
                              hipStream_t stream) {
    const float* x       = (const float*)d_in[0];
    const int*   idx_a   = (const int*)d_in[1];
    const int*   idx_b   = (const int*)d_in[2];
    const float* weights = (const float*)d_in[3];
    float*       out     = (float*)d_out;
    float4*      coeffs  = (float4*)d_ws;   // OUT_DIM * 16 B = 128 KiB scratch

    (void)in_sizes; (void)n_in; (void)out_size; (void)ws_size;

    gate_coeffs_kernel<<<(OUT_DIM + TPB - 1) / TPB, TPB, 0, stream>>>(weights, coeffs);
    logic_layer_kernel<<<BATCH / ROWS, TPB, 0, stream>>>(x, idx_a, idx_b, coeffs, out);
}